// SimpleGNNWithAttention_62199716380682
// MI455X (gfx1250) — compile-verified
//
#include <hip/hip_runtime.h>
#include <math.h>

// ---------------------------------------------------------------------------
// GAT (2 layers, H=32 collapsed via rank-1 algebra) + final 30x10 linear head.
// Graph part: scalar segment softmax with native gfx1250 atomics.
// Final GEMM: V_WMMA_F32_16X16X4_F32 on zero-padded operands (no predication
// in the inner loop; uniform-guarded store epilogue).
// ---------------------------------------------------------------------------

typedef float v2f __attribute__((ext_vector_type(2)));
typedef float v8f __attribute__((ext_vector_type(8)));

#define TPB 256
#define NSLOPE 0.2f

// Order-preserving float<->uint mapping for integer atomicMax on scores.
__device__ __forceinline__ unsigned f2o(float f) {
  unsigned u = __float_as_uint(f);
  return (u & 0x80000000u) ? ~u : (u | 0x80000000u);
}
__device__ __forceinline__ float o2f(unsigned u) {
  return __uint_as_float((u & 0x80000000u) ? (u & 0x7FFFFFFFu) : ~u);
}

// -------- init per-layer segment state (mU = f2o(-inf) = 0x007FFFFF) --------
__global__ void k_seg_init(unsigned* __restrict__ mU, float* __restrict__ denom,
                           float* __restrict__ num, int N) {
  int i = blockIdx.x * blockDim.x + threadIdx.x;
  if (i < N) {
    mU[i] = 0x007FFFFFu;
    denom[i] = 0.0f;
    num[i] = 0.0f;
  }
}

// -------- layer-1 node pre-pass: e_src = x * <W1,a_src1>, e_dst likewise ----
__global__ void k_nodes_l1(const float* __restrict__ x, const float* __restrict__ W1,
                           const float* __restrict__ a_src1, const float* __restrict__ a_dst1,
                           float* __restrict__ eS, float* __restrict__ eD, int N, int H) {
  int i = blockIdx.x * blockDim.x + threadIdx.x;
  if (i >= N) return;
  float cs = 0.0f, cd = 0.0f;
  #pragma unroll 8
  for (int j = 0; j < H; ++j) {   // uniform loads -> scalar cache
    float w = W1[j];
    cs = fmaf(w, a_src1[j], cs);
    cd = fmaf(w, a_dst1[j], cd);
  }
  float xv = x[i];
  eS[i] = xv * cs;
  eD[i] = xv * cd;
}

// -------- edge pass 1: leaky-relu score, store, segment max ------------------
__global__ void k_edge_max(const int* __restrict__ src, const int* __restrict__ dst,
                           int E, int E2,
                           const float* __restrict__ eS, const float* __restrict__ eD,
                           float* __restrict__ scores, unsigned* __restrict__ mU) {
  int i = blockIdx.x * blockDim.x + threadIdx.x;
  if (i >= E2) return;
  int s, d;
  if (i < E) { s = src[i]; d = dst[i]; } else { s = i - E; d = s; }  // self loops
  float sc = eS[s] + eD[d];
  sc = (sc > 0.0f) ? sc : NSLOPE * sc;
  scores[i] = sc;
  atomicMax(&mU[d], f2o(sc));
}

// -------- edge pass 2: ex = exp(score - m[d]); denom += ex; num += ex*feat[s]
__global__ void k_edge_acc(const int* __restrict__ src, const int* __restrict__ dst,
                           int E, int E2,
                           const float* __restrict__ scores, const unsigned* __restrict__ mU,
                           const float* __restrict__ feat,
                           float* __restrict__ denom, float* __restrict__ num) {
  int i = blockIdx.x * blockDim.x + threadIdx.x;
  if (i >= E2) return;
  int s, d;
  if (i < E) { s = src[i]; d = dst[i]; } else { s = i - E; d = s; }
  float ex = __expf(scores[i] - o2f(mU[d]));
  unsafeAtomicAdd(&denom[d], ex);                 // global_atomic_add_f32
  unsafeAtomicAdd(&num[d], ex * feat[s]);
}

// -------- between layers: s1 = num/denom; h1=relu(s1*W1+b1); h2=h1.W2 -------
__global__ void k_mid(const float* __restrict__ denom, const float* __restrict__ num,
                      const float* __restrict__ W1, const float* __restrict__ b1,
                      const float* __restrict__ W2,
                      const float* __restrict__ a_src2, const float* __restrict__ a_dst2,
                      float* __restrict__ h2, float* __restrict__ eS, float* __restrict__ eD,
                      int N, int H) {
  int i = blockIdx.x * blockDim.x + threadIdx.x;
  if (i >= N) return;
  float s1 = num[i] / denom[i];
  float acc = 0.0f;
  #pragma unroll 8
  for (int j = 0; j < H; ++j) {
    float hj = fmaf(s1, W1[j], b1[j]);
    hj = fmaxf(hj, 0.0f);                    // relu after layer 1
    acc = fmaf(hj, W2[j], acc);
  }
  h2[i] = acc;
  eS[i] = acc * a_src2[0];
  eD[i] = acc * a_dst2[0];
}

// -------- layer-2 finalize: o2 = relu(num/denom + b2) ------------------------
__global__ void k_final_nodes(const float* __restrict__ denom, const float* __restrict__ num,
                              const float* __restrict__ b2, float* __restrict__ o2, int N) {
  int i = blockIdx.x * blockDim.x + threadIdx.x;
  if (i < N) o2[i] = fmaxf(num[i] / denom[i] + b2[0], 0.0f);
}

// -------- pack A: o2 rows [rows,30] -> [padRows,32] zero-padded -------------
__global__ void k_pack_A(const float* __restrict__ o2, float* __restrict__ Apad,
                         int rows, int padRows) {
  int idx = blockIdx.x * blockDim.x + threadIdx.x;
  if (idx >= padRows * 32) return;
  int r = idx >> 5;
  int k = idx & 31;
  Apad[idx] = (r < rows && k < 30) ? o2[r * 30 + k] : 0.0f;
}

// -------- pack B: Wlin [30,10] -> transposed padded Bt[16][32] --------------
__global__ void k_pack_B(const float* __restrict__ Wlin, float* __restrict__ Bt) {
  int idx = blockIdx.x * blockDim.x + threadIdx.x;
  if (idx >= 16 * 32) return;
  int n = idx >> 5;
  int k = idx & 31;
  Bt[idx] = (n < 10 && k < 30) ? Wlin[k * 10 + n] : 0.0f;
}

// -------- final linear head: [rows,30] @ [30,10] + blin via WMMA f32 --------
// One wave32 per 16x16 output tile; K padded 30 -> 32 (8 x WMMA 16x16x4).
// A layout (16x4 f32): lane L -> M=L&15; lanes<16 hold K={0,1}, lanes>=16 K={2,3}.
// B layout mirrors A with N in place of M. All inner-loop loads unconditional
// 8B-aligned b64 (operands pre-padded). Store epilogue: single uniform branch
// for full tiles -> 8 unguarded stores; per-row predication only on tail tile.
__global__ void k_lin_wmma(const float* __restrict__ Apad, const float* __restrict__ Bt,
                           const float* __restrict__ blin, float* __restrict__ out,
                           int rows) {
  const int tile = blockIdx.x;
  const int lane = threadIdx.x;         // 0..31
  const int row0 = tile * 16;
  const int m    = lane & 15;
  const int n    = lane & 15;
  const int koff = (lane >> 4) << 1;    // 0 for lanes 0-15, 2 for lanes 16-31

  const float* arow = Apad + (size_t)(row0 + m) * 32 + koff;
  const float* brow = Bt   + (size_t)n * 32 + koff;

  v8f c = {};
  #pragma unroll
  for (int k0 = 0; k0 < 32; k0 += 4) {
    v2f a = *(const v2f*)(arow + k0);   // 8B-aligned: koff,k0 both even
    v2f b = *(const v2f*)(brow + k0);
    // (neg_a, A, neg_b, B, c_mod, C, reuse_a, reuse_b)
    c = __builtin_amdgcn_wmma_f32_16x16x4_f32(false, a, false, b, (short)0, c,
                                              false, false);
  }

  const int mbase = (lane >> 4) << 3;   // 0 or 8
  if (n < 10) {
    const float bb = blin[n];
    float* po = out + (size_t)(row0 + mbase) * 10 + n;
    if (row0 + 16 <= rows) {            // uniform: every tile except a tail
      #pragma unroll
      for (int r = 0; r < 8; ++r) po[r * 10] = c[r] + bb;
    } else {
      #pragma unroll
      for (int r = 0; r < 8; ++r)
        if (row0 + mbase + r < rows) po[r * 10] = c[r] + bb;
    }
  }
}

// ---------------------------------------------------------------------------
extern "C" void kernel_launch(void* const* d_in, const int* in_sizes, int n_in,
                              void* d_out, int out_size, void* d_ws, size_t ws_size,
                              hipStream_t stream) {
  const float* x      = (const float*)d_in[0];
  const int*   ei     = (const int*)  d_in[1];   // [2, E] row-major
  const float* W1     = (const float*)d_in[2];
  const float* a_src1 = (const float*)d_in[3];
  const float* a_dst1 = (const float*)d_in[4];
  const float* b1     = (const float*)d_in[5];
  const float* W2     = (const float*)d_in[6];
  const float* a_src2 = (const float*)d_in[7];
  const float* a_dst2 = (const float*)d_in[8];
  const float* b2     = (const float*)d_in[9];
  const float* Wlin   = (const float*)d_in[10];
  const float* blin   = (const float*)d_in[11];
  float* out = (float*)d_out;

  const int N  = in_sizes[0];            // 480000
  const int H  = in_sizes[2];            // 32
  const int E  = in_sizes[1] / 2;        // 7,680,000
  const int E2 = E + N;                  // + self loops
  const int* src = ei;
  const int* dst = ei + E;
  const int rows  = N / 30;              // 16000
  const int tiles = (rows + 15) / 16;    // 1000
  const int padRows = tiles * 16;

  // Workspace carve-out (all buffers written before read; reused across layers)
  char* ws = (char*)d_ws;
  size_t off = 0;
  auto carve = [&](size_t bytes) -> void* {
    void* p = ws + off;
    off += (bytes + 255) & ~(size_t)255;
    return p;
  };
  float*    eS     = (float*)   carve((size_t)N * 4);
  float*    eD     = (float*)   carve((size_t)N * 4);
  unsigned* mU     = (unsigned*)carve((size_t)N * 4);
  float*    denom  = (float*)   carve((size_t)N * 4);
  float*    num    = (float*)   carve((size_t)N * 4);
  float*    h2     = (float*)   carve((size_t)N * 4);
  float*    o2     = (float*)   carve((size_t)N * 4);
  float*    Apad   = (float*)   carve((size_t)padRows * 32 * 4);
  float*    Bt     = (float*)   carve((size_t)16 * 32 * 4);
  float*    scores = (float*)   carve((size_t)E2 * 4);
  (void)ws_size; (void)n_in; (void)out_size;

  const int nbN = (N  + TPB - 1) / TPB;
  const int nbE = (E2 + TPB - 1) / TPB;

  // ---- Layer 1 (feature aggregation collapsed to scalar via rank-1 W1) ----
  k_seg_init <<<nbN, TPB, 0, stream>>>(mU, denom, num, N);
  k_nodes_l1 <<<nbN, TPB, 0, stream>>>(x, W1, a_src1, a_dst1, eS, eD, N, H);
  k_edge_max <<<nbE, TPB, 0, stream>>>(src, dst, E, E2, eS, eD, scores, mU);
  k_edge_acc <<<nbE, TPB, 0, stream>>>(src, dst, E, E2, scores, mU, x, denom, num);

  // ---- Node transform between layers (relu, H-dim dot, layer-2 scores) ----
  k_mid      <<<nbN, TPB, 0, stream>>>(denom, num, W1, b1, W2, a_src2, a_dst2,
                                       h2, eS, eD, N, H);

  // ---- Layer 2 (scalar features) ----
  k_seg_init <<<nbN, TPB, 0, stream>>>(mU, denom, num, N);
  k_edge_max <<<nbE, TPB, 0, stream>>>(src, dst, E, E2, eS, eD, scores, mU);
  k_edge_acc <<<nbE, TPB, 0, stream>>>(src, dst, E, E2, scores, mU, h2, denom, num);
  k_final_nodes<<<nbN, TPB, 0, stream>>>(denom, num, b2, o2, N);

  // ---- Pack operands, then final linear head on the tensor core path ----
  k_pack_A <<<(padRows * 32 + TPB - 1) / TPB, TPB, 0, stream>>>(o2, Apad, rows, padRows);
  k_pack_B <<<(16 * 32 + TPB - 1) / TPB, TPB, 0, stream>>>(Wlin, Bt);
  k_lin_wmma <<<tiles, 32, 0, stream>>>(Apad, Bt, blin, out, rows);
}